// GMMLoss_48241072669053
// MI455X (gfx1250) — compile-verified
//
#include <hip/hip_runtime.h>
#include <math.h>

// GMM negative log-likelihood on MI455X (gfx1250).
// features: [16,16,7,64,64] f32 -> [256][7][4096]
// targets : [16,16,32,4]    f32 -> [256][32][4]  (mask, tv0, tv1, tv2)
// out     : [16,16] f32 -> [256]
//
// t(s,p) = x_s . y_p  (K=7, padded to 8)
//   x = [tv0,tv1,tv2, tv0^2,tv1^2,tv2^2, 1, 0]
//   y = [B0,B1,B2, -C0,-C1,-C2, log(prob)-A_p, 0]
//   B_g = mu/sigma^2, C_g = 1/(2 sigma^2), A_p = sum_g (mu^2 C_g + log sigma)
// out[b] = -sum_s mask_s * ( logsumexp_p t(s,p) - log(sum_p prob_p) )
//
// One workgroup (8 wave32) per batch; V_WMMA_F32_16X16X4_F32 computes the
// 32x7x4096 GEMM in 16x16 tiles; streaming logsumexp in registers.

typedef __attribute__((ext_vector_type(2))) float v2f;
typedef __attribute__((ext_vector_type(8))) float v8f;

#define NPIX   4096
#define NSPOT  32
#define WAVES  8
#define THREADS 256
#define NEG_BIG (-3.402823466e38f)

__global__ __launch_bounds__(THREADS)
void gmm_loss_wmma_kernel(const float* __restrict__ features,
                          const float* __restrict__ targets,
                          float* __restrict__ out)
{
    const int b    = blockIdx.x;
    const int tid  = threadIdx.x;
    const int lane = tid & 31;
    const int wave = tid >> 5;
    const bool hi  = lane >= 16;   // upper half-wave
    const int  l16 = lane & 15;

    const float* feat = features + (size_t)b * 7 * NPIX;
    const float* tgt  = targets  + (size_t)b * NSPOT * 4;

    __shared__ float red[THREADS];
    __shared__ float pm[WAVES][NSPOT];
    __shared__ float ps[WAVES][NSPOT];
    __shared__ float sh_logS;

    // ---------- Phase 1: logS = log( sum_p clip(prob_p, 1e-20) ) ----------
    float accp = 0.0f;
    for (int p = tid; p < NPIX; p += THREADS)
        accp += fmaxf(feat[p], 1e-20f);
    red[tid] = accp;
    __syncthreads();
    #pragma unroll
    for (int off = THREADS / 2; off > 0; off >>= 1) {
        if (tid < off) red[tid] += red[tid + off];
        __syncthreads();
    }
    if (tid == 0) sh_logS = __logf(red[0]);
    __syncthreads();
    const float logS = sh_logS;

    // ---------- A fragments (spot matrix), constant over pixel tiles ----------
    // ISA 32-bit A 16x4 layout: VGPR0 = K0 (lanes 0-15) | K2 (lanes 16-31),
    //                           VGPR1 = K1 | K3.
    v2f afrag[2][2];   // [Mtile][Kstep]
    #pragma unroll
    for (int mt = 0; mt < 2; ++mt) {
        const int s = l16 + 16 * mt;
        const float tv0 = tgt[4 * s + 1];
        const float tv1 = tgt[4 * s + 2];
        const float tv2 = tgt[4 * s + 3];
        // Kstep0: k = 0..3 -> [tv0, tv1, tv2, tv0^2]
        afrag[mt][0][0] = hi ? tv2         : tv0;
        afrag[mt][0][1] = hi ? (tv0 * tv0) : tv1;
        // Kstep1: k = 4..7 -> [tv1^2, tv2^2, 1, 0]
        afrag[mt][1][0] = hi ? 1.0f        : (tv1 * tv1);
        afrag[mt][1][1] = hi ? 0.0f        : (tv2 * tv2);
    }

    // ---------- streaming logsumexp accumulators ----------
    // D tile layout: VGPR v -> row M=v (lanes 0-15) / M=v+8 (lanes 16-31)
    float rm[2][8], rs[2][8];
    #pragma unroll
    for (int mt = 0; mt < 2; ++mt)
        #pragma unroll
        for (int v = 0; v < 8; ++v) { rm[mt][v] = NEG_BIG; rs[mt][v] = 0.0f; }

    // ---------- Phase 2: pixel tiles (uniform per wave; EXEC all-ones) ----------
    for (int tile = wave; tile < NPIX / 16; tile += WAVES) {
        const int p = tile * 16 + l16;   // both half-waves load the same 16 px

        const float prob = fmaxf(feat[p], 1e-20f);
        const float mu0  = feat[1 * NPIX + p];
        const float mu1  = feat[2 * NPIX + p];
        const float mu2  = feat[3 * NPIX + p];
        const float sg0  = fmaxf(feat[4 * NPIX + p], 1e-10f);
        const float sg1  = fmaxf(feat[5 * NPIX + p], 1e-10f);
        const float sg2  = fmaxf(feat[6 * NPIX + p], 1e-10f);

        const float c0 = 0.5f / (sg0 * sg0);
        const float c1 = 0.5f / (sg1 * sg1);
        const float c2 = 0.5f / (sg2 * sg2);
        const float b0 = 2.0f * mu0 * c0;
        const float b1 = 2.0f * mu1 * c1;
        const float b2 = 2.0f * mu2 * c2;
        const float ap = mu0 * mu0 * c0 + mu1 * mu1 * c1 + mu2 * mu2 * c2
                       + __logf(sg0) + __logf(sg1) + __logf(sg2);
        const float y6 = __logf(prob) - ap;

        // B 4x16 fragments, same K-to-VGPR striping as A:
        v2f bfrag0, bfrag1;
        bfrag0[0] = hi ? b2   : b0;    // K=2 | K=0
        bfrag0[1] = hi ? -c0  : b1;    // K=3 | K=1
        bfrag1[0] = hi ? y6   : -c1;   // K=6 | K=4
        bfrag1[1] = hi ? 0.0f : -c2;   // K=7 | K=5

        #pragma unroll
        for (int mt = 0; mt < 2; ++mt) {
            v8f d = {};
            d = __builtin_amdgcn_wmma_f32_16x16x4_f32(
                    false, afrag[mt][1], false, bfrag1, (short)0, d, false, false);
            d = __builtin_amdgcn_wmma_f32_16x16x4_f32(
                    false, afrag[mt][0], false, bfrag0, (short)0, d, false, false);
            #pragma unroll
            for (int v = 0; v < 8; ++v) {
                const float t  = d[v];
                const float mn = fmaxf(rm[mt][v], t);
                rs[mt][v] = rs[mt][v] * __expf(rm[mt][v] - mn) + __expf(t - mn);
                rm[mt][v] = mn;
            }
        }
    }

    // ---------- reduce logsumexp partials across 16-lane groups ----------
    #pragma unroll
    for (int mt = 0; mt < 2; ++mt) {
        #pragma unroll
        for (int v = 0; v < 8; ++v) {
            float m = rm[mt][v], s = rs[mt][v];
            #pragma unroll
            for (int k = 1; k < 16; k <<= 1) {
                const float mo = __shfl_xor(m, k, 32);
                const float so = __shfl_xor(s, k, 32);
                const float M  = fmaxf(m, mo);
                s = s * __expf(m - M) + so * __expf(mo - M);
                m = M;
            }
            if (l16 == 0) {
                const int spot = (hi ? v + 8 : v) + 16 * mt;
                pm[wave][spot] = m;
                ps[wave][spot] = s;
            }
        }
    }
    __syncthreads();

    // ---------- combine across waves, apply mask, reduce over spots ----------
    if (wave == 0) {
        const int spot = lane;
        float m = pm[0][spot], s = ps[0][spot];
        #pragma unroll
        for (int w = 1; w < WAVES; ++w) {
            const float mo = pm[w][spot];
            const float so = ps[w][spot];
            const float M  = fmaxf(m, mo);
            s = s * __expf(m - M) + so * __expf(mo - M);
            m = M;
        }
        const float log_gmm = m + __logf(s) - logS;
        float val = tgt[4 * spot] * log_gmm;   // mask * logprob
        #pragma unroll
        for (int k = 1; k < 32; k <<= 1)
            val += __shfl_xor(val, k, 32);
        if (lane == 0) out[b] = -val;
    }
}

extern "C" void kernel_launch(void* const* d_in, const int* in_sizes, int n_in,
                              void* d_out, int out_size, void* d_ws, size_t ws_size,
                              hipStream_t stream) {
    (void)in_sizes; (void)n_in; (void)d_ws; (void)ws_size; (void)out_size;
    const float* features = (const float*)d_in[0];
    const float* targets  = (const float*)d_in[1];
    float* out            = (float*)d_out;
    gmm_loss_wmma_kernel<<<dim3(256), dim3(THREADS), 0, stream>>>(features, targets, out);
}